// PraxisAttention_67542655696922
// MI455X (gfx1250) — compile-verified
//
#include <hip/hip_runtime.h>

typedef __attribute__((ext_vector_type(2))) float v2f;
typedef __attribute__((ext_vector_type(8))) float v8f;

#define B_    4
#define H_    16
#define L_    4096
#define D_    64
#define BH_   (B_ * H_)
#define C_    64            // chunks per (b,h)
#define ROWS_ (L_ / C_)     // 64 rows per chunk
#define EPS_  1e-6f

__device__ __forceinline__ float elu1(float x) {
    // elu(x)+1 : x>0 -> x+1 ; else exp(x)
    return x > 0.0f ? x + 1.0f : expf(x);
}

// ---------------------------------------------------------------------------
// Pass 1: per-chunk channel sums of kf and kf*v (with mask), 1 wave per chunk.
// ws layout per (bh,c): [0..63] = sum(kf), [64..127] = sum(kf*v*m^2)
// ---------------------------------------------------------------------------
__global__ void __launch_bounds__(32) chunk_partials(
    const float* __restrict__ k, const float* __restrict__ v,
    const float* __restrict__ mask, float* __restrict__ ws)
{
    const int gid  = blockIdx.x;          // bh*C_ + c
    const int c    = gid % C_;
    const int bh   = gid / C_;
    const int b    = bh / H_;
    const int lane = threadIdx.x;
    const int d0   = lane * 2;
    const int l0   = c * ROWS_;

    const float* kp = k + ((size_t)bh * L_ + l0) * D_ + d0;
    const float* vp = v + ((size_t)bh * L_ + l0) * D_ + d0;
    const float* mp = mask + (size_t)b * L_ + l0;

    float sk0 = 0.f, sk1 = 0.f, sv0 = 0.f, sv1 = 0.f;
    #pragma unroll 4
    for (int r = 0; r < ROWS_; ++r) {
        v2f k2 = *(const v2f*)(kp + (size_t)r * D_);
        v2f v2 = *(const v2f*)(vp + (size_t)r * D_);
        float m  = mp[r];
        float f0 = elu1(k2.x) * m, f1 = elu1(k2.y) * m;
        sk0 += f0;              sk1 += f1;
        sv0 += f0 * (v2.x * m); sv1 += f1 * (v2.y * m);
    }
    float* w = ws + (size_t)gid * 128;
    w[d0]      = sk0;  w[d0 + 1]      = sk1;
    w[64 + d0] = sv0;  w[64 + d0 + 1] = sv1;
}

// ---------------------------------------------------------------------------
// Pass 2: in-place exclusive scan of chunk sums along c, per (b,h).
// ---------------------------------------------------------------------------
__global__ void __launch_bounds__(32) chunk_scan(float* __restrict__ ws)
{
    const int bh   = blockIdx.x;
    const int lane = threadIdx.x;
    const int d0   = lane * 2;
    float rk0 = 0.f, rk1 = 0.f, rv0 = 0.f, rv1 = 0.f;
    for (int c = 0; c < C_; ++c) {
        float* w = ws + ((size_t)bh * C_ + c) * 128;
        float ck0 = w[d0],      ck1 = w[d0 + 1];
        float cv0 = w[64 + d0], cv1 = w[64 + d0 + 1];
        w[d0]      = rk0; w[d0 + 1]      = rk1;
        w[64 + d0] = rv0; w[64 + d0 + 1] = rv1;
        rk0 += ck0; rk1 += ck1; rv0 += cv0; rv1 += cv1;
    }
}

// ---------------------------------------------------------------------------
// Pass 3: per chunk (1 wave): intra-tile inclusive cumsum via triangular
// matmul on V_WMMA_F32_16X16X4_F32, z row-dot via lane reduction, output.
// ---------------------------------------------------------------------------
__global__ void __launch_bounds__(32) linattn_main(
    const float* __restrict__ q, const float* __restrict__ k,
    const float* __restrict__ v, const float* __restrict__ mask,
    const float* __restrict__ ws, float* __restrict__ out)
{
    const int gid  = blockIdx.x;
    const int c    = gid % C_;
    const int bh   = gid / C_;
    const int b    = bh / H_;
    const int lane = threadIdx.x;
    const int half = (lane >= 16) ? 1 : 0;  // C-layout row half
    const int m15  = lane & 15;

    // Triangular A slices (16x4 each), constant across tiles/groups.
    // A layout: lanes 0-15 hold M=0..15 {K=0,K=1}; lanes 16-31 hold {K=2,K=3}.
    v2f A[4];
    #pragma unroll
    for (int s = 0; s < 4; ++s) {
        const int klo = 4 * s + half * 2;
        A[s].x = (klo     <= m15) ? 1.0f : 0.0f;
        A[s].y = (klo + 1 <= m15) ? 1.0f : 0.0f;
    }

    // Exclusive cross-chunk prefixes, one per N-group, in C-layout (per-channel).
    const float* w = ws + (size_t)gid * 128;
    float pk[4], pkv[4];
    #pragma unroll
    for (int g = 0; g < 4; ++g) {
        pk[g]  = w[g * 16 + m15];
        pkv[g] = w[64 + g * 16 + m15];
    }

    const size_t rowbase = (size_t)bh * L_;
    const float* mp = mask + (size_t)b * L_;
    const int l0 = c * ROWS_;

    for (int t = 0; t < 4; ++t) {            // 4 tiles of 16 rows
        const int tb = l0 + t * 16;
        v8f zp = {};                         // z partials for rows tb+half*8+r
        v8f ckv_g[4], qf_g[4];

        #pragma unroll
        for (int g = 0; g < 4; ++g) {
            const int d = g * 16 + m15;
            // Seed accumulators with the running prefix (row-independent).
            v8f ck, ckv;
            #pragma unroll
            for (int i = 0; i < 8; ++i) { ck[i] = pk[g]; ckv[i] = pkv[g]; }

            #pragma unroll
            for (int s = 0; s < 4; ++s) {
                // B layout: lanes 0-15 rows {4s,4s+1}; lanes 16-31 rows {4s+2,4s+3}
                const int r0 = tb + 4 * s + half * 2;
                const int r1 = r0 + 1;
                const float m0 = mp[r0], m1 = mp[r1];
                const float kb0 = elu1(k[(rowbase + r0) * D_ + d]) * m0;
                const float kb1 = elu1(k[(rowbase + r1) * D_ + d]) * m1;
                const float vb0 = v[(rowbase + r0) * D_ + d] * m0;
                const float vb1 = v[(rowbase + r1) * D_ + d] * m1;
                v2f Bk  = {kb0, kb1};
                v2f Bkv = {kb0 * vb0, kb1 * vb1};
                ck  = __builtin_amdgcn_wmma_f32_16x16x4_f32(
                          false, A[s], false, Bk,  (short)0, ck,  false, false);
                ckv = __builtin_amdgcn_wmma_f32_16x16x4_f32(
                          false, A[s], false, Bkv, (short)0, ckv, false, false);
            }

            // Carry: inclusive value at local row 15 (C VGPR7, upper half).
            pk[g]  = __shfl(ck[7],  16 + m15, 32);
            pkv[g] = __shfl(ckv[7], 16 + m15, 32);

            // qf in C-layout; accumulate z partials.
            v8f qf;
            #pragma unroll
            for (int r = 0; r < 8; ++r) {
                const int row = tb + half * 8 + r;
                qf[r] = elu1(q[(rowbase + row) * D_ + d]);
            }
            #pragma unroll
            for (int r = 0; r < 8; ++r) zp[r] += qf[r] * ck[r];
            ckv_g[g] = ckv;
            qf_g[g]  = qf;
        }

        // Reduce z over the 16 lanes of each half (rows differ between halves).
        #pragma unroll
        for (int r = 0; r < 8; ++r) {
            float s = zp[r];
            s += __shfl_xor(s, 1, 32);
            s += __shfl_xor(s, 2, 32);
            s += __shfl_xor(s, 4, 32);
            s += __shfl_xor(s, 8, 32);
            zp[r] = s;
        }

        // Normalize and store.
        #pragma unroll
        for (int r = 0; r < 8; ++r) {
            const int row = tb + half * 8 + r;
            const float z   = (zp[r] + EPS_) * mp[row];
            const float inv = 1.0f / z;
            #pragma unroll
            for (int g = 0; g < 4; ++g) {
                const int d = g * 16 + m15;
                out[(rowbase + row) * D_ + d] = qf_g[g][r] * ckv_g[g][r] * inv;
            }
        }
    }
}

// ---------------------------------------------------------------------------
extern "C" void kernel_launch(void* const* d_in, const int* in_sizes, int n_in,
                              void* d_out, int out_size, void* d_ws, size_t ws_size,
                              hipStream_t stream)
{
    const float* q    = (const float*)d_in[0];
    const float* k    = (const float*)d_in[1];
    const float* v    = (const float*)d_in[2];
    const float* mask = (const float*)d_in[3];
    float* out = (float*)d_out;
    float* ws  = (float*)d_ws;   // needs BH_*C_*128 floats = 2 MB

    chunk_partials<<<BH_ * C_, 32, 0, stream>>>(k, v, mask, ws);
    chunk_scan    <<<BH_,      32, 0, stream>>>(ws);
    linattn_main  <<<BH_ * C_, 32, 0, stream>>>(q, k, v, mask, ws, out);
}